// FocalLoss_55645596287615
// MI455X (gfx1250) — compile-verified
//
#include <hip/hip_runtime.h>

// ---------------------------------------------------------------------------
// Focal loss (RetinaNet) for MI455X / gfx1250.
// Bandwidth-bound: one fused streaming pass over classifications (154 MB),
// deterministic two-stage reduction (block partials -> fixed-tree finalize).
// Wave-level sums use V_WMMA_F32_16X16X4_F32 with an all-ones B matrix
// (exact f32; relies only on the documented A and C/D VGPR layouts).
// ---------------------------------------------------------------------------

typedef __attribute__((ext_vector_type(2))) float v2f;
typedef __attribute__((ext_vector_type(8))) float v8f;

#define TPB 256
#define MAXM 64
#define ALPHA_F 0.25f
#define PCLIP_LO 1e-4f
#define PCLIP_HI 0.9999f

// Exact f32 wave(32) sum via WMMA:
// A[16x4]: lane l<16 owns A[l][0] (VGPR0) & A[l][1] (VGPR1);
//          lane l>=16 owns A[l-16][2] & A[l-16][3].
// Put v in VGPR0, 0 in VGPR1; B = ones => D[m][n] = v_m + v_{m+16} for all n.
// C/D layout: lane n<16 VGPR j = D[j][n]; lane n+16 VGPR j = D[j+8][n].
// sum of 8 C regs + shfl_xor(16) => full 32-lane sum on every lane.
__device__ __forceinline__ float wave_sum_wmma(float v) {
  v2f a;    a[0] = v;    a[1] = 0.0f;
  v2f ones; ones[0] = 1.0f; ones[1] = 1.0f;
  v8f c = {};
  c = __builtin_amdgcn_wmma_f32_16x16x4_f32(false, a, false, ones,
                                            (short)0, c, false, false);
  float s = c[0] + c[1] + c[2] + c[3] + c[4] + c[5] + c[6] + c[7];
  s += __shfl_xor(s, 16, 32);
  return s;
}

template <int CT, int MT>
__global__ __launch_bounds__(TPB) void focal_main_kernel(
    const float* __restrict__ cls_p,   // [B, A, C]
    const float* __restrict__ regr,    // [B, A, 4]
    const float* __restrict__ anch,    // [A, 4] (anchors[0])
    const float* __restrict__ ann,     // [B, M, 5]
    float* __restrict__ ws_cls, float* __restrict__ ws_reg,
    float* __restrict__ ws_np,
    int A, int Crt, int Mrt) {
  const int C = CT ? CT : Crt;
  const int M = MT ? MT : Mrt;
  const int b = blockIdx.y;
  const int a = blockIdx.x * TPB + threadIdx.x;

  __shared__ float gt[MAXM * 5];
  __shared__ float red_c[TPB / 32], red_r[TPB / 32], red_n[TPB / 32];

  for (int i = threadIdx.x; i < M * 5; i += TPB)
    gt[i] = ann[(size_t)b * M * 5 + i];
  __syncthreads();

  float cls_acc = 0.0f, reg_acc = 0.0f;
  int is_pos = 0;

  if (a < A) {
    const float4 ab = *(const float4*)(anch + (size_t)a * 4);
    const float aw = ab.z - ab.x, ah = ab.w - ab.y;
    const float acx = ab.x + 0.5f * aw, acy = ab.y + 0.5f * ah;
    const float aarea = aw * ah;

    // IoU max / argmax over M GT boxes (first-max wins, matching jnp.argmax)
    float best = -3.402823466e38f;
    int arg = 0;
#pragma unroll 8
    for (int m = 0; m < M; ++m) {
      const float gx1 = gt[m * 5 + 0], gy1 = gt[m * 5 + 1];
      const float gx2 = gt[m * 5 + 2], gy2 = gt[m * 5 + 3];
      const float gl  = gt[m * 5 + 4];
      const float iw = fmaxf(fminf(ab.z, gx2) - fmaxf(ab.x, gx1), 0.0f);
      const float ih = fmaxf(fminf(ab.w, gy2) - fmaxf(ab.y, gy1), 0.0f);
      const float inter = iw * ih;
      const float ua = fmaxf(aarea + (gx2 - gx1) * (gy2 - gy1) - inter, 1e-8f);
      const float iou = (gl != -1.0f) ? (inter / ua) : -1.0f;
      if (iou > best) { best = iou; arg = m; }
    }
    const bool pos = (best >= 0.5f);
    const bool neg = (best < 0.4f);
    is_pos = pos ? 1 : 0;

    if (pos | neg) {
      const size_t base = ((size_t)b * A + a) * C;
      float s = 0.0f;
      if (CT > 0) {  // compile-time C (multiple of 4): vectorized stream
        const float4* pv = (const float4*)(cls_p + base);
#pragma unroll 4
        for (int i = 0; i < C / 4; ++i) {
          const float4 v = pv[i];
          float p0 = fminf(fmaxf(v.x, PCLIP_LO), PCLIP_HI);
          float p1 = fminf(fmaxf(v.y, PCLIP_LO), PCLIP_HI);
          float p2 = fminf(fmaxf(v.z, PCLIP_LO), PCLIP_HI);
          float p3 = fminf(fmaxf(v.w, PCLIP_LO), PCLIP_HI);
          // t==0 branch: (1-alpha) * p^2 * (-log(1-p))
          s += (1.0f - ALPHA_F) * p0 * p0 * (-__logf(1.0f - p0));
          s += (1.0f - ALPHA_F) * p1 * p1 * (-__logf(1.0f - p1));
          s += (1.0f - ALPHA_F) * p2 * p2 * (-__logf(1.0f - p2));
          s += (1.0f - ALPHA_F) * p3 * p3 * (-__logf(1.0f - p3));
        }
      } else {  // generic fallback
        for (int cc = 0; cc < C; ++cc) {
          float p = fminf(fmaxf(cls_p[base + cc], PCLIP_LO), PCLIP_HI);
          s += (1.0f - ALPHA_F) * p * p * (-__logf(1.0f - p));
        }
      }
      if (pos) {
        // fix up the one-hot class: replace neg term with pos term
        const float glab = gt[arg * 5 + 4];
        const int label = (int)fminf(fmaxf(glab, 0.0f), (float)(C - 1));
        float pl = fminf(fmaxf(cls_p[base + label], PCLIP_LO), PCLIP_HI);
        const float negt = (1.0f - ALPHA_F) * pl * pl * (-__logf(1.0f - pl));
        const float post = ALPHA_F * (1.0f - pl) * (1.0f - pl) * (-__logf(pl));
        s += post - negt;

        // regression L1 vs box-encoded targets
        const float gx1 = gt[arg * 5 + 0], gy1 = gt[arg * 5 + 1];
        const float gx2 = gt[arg * 5 + 2], gy2 = gt[arg * 5 + 3];
        const float gw = fmaxf(gx2 - gx1, 1.0f), gh = fmaxf(gy2 - gy1, 1.0f);
        const float gcx = gx1 + 0.5f * (gx2 - gx1);
        const float gcy = gy1 + 0.5f * (gy2 - gy1);
        const float t0 = (gcx - acx) / aw;
        const float t1 = (gcy - acy) / ah;
        const float t2 = __logf(gw / aw);
        const float t3 = __logf(gh / ah);
        const float4 rv = *(const float4*)(regr + ((size_t)b * A + a) * 4);
        reg_acc = fabsf(rv.x - t0) + fabsf(rv.y - t1) +
                  fabsf(rv.z - t2) + fabsf(rv.w - t3);
      }
      cls_acc = s;
    }
  }

  // --- block reduction; all threads converged here (EXEC all-ones for WMMA)
  const float wc = wave_sum_wmma(cls_acc);
  const float wr = wave_sum_wmma(reg_acc);
  const unsigned long long bal = __ballot(is_pos);
  const float wn = (float)__popcll(bal);

  const int wid = threadIdx.x >> 5;
  if ((threadIdx.x & 31) == 0) { red_c[wid] = wc; red_r[wid] = wr; red_n[wid] = wn; }
  __syncthreads();
  if (threadIdx.x == 0) {
    float c = 0.0f, r = 0.0f, n = 0.0f;
#pragma unroll
    for (int w = 0; w < TPB / 32; ++w) { c += red_c[w]; r += red_r[w]; n += red_n[w]; }
    const size_t o = (size_t)b * gridDim.x + blockIdx.x;
    ws_cls[o] = c; ws_reg[o] = r; ws_np[o] = n;
  }
}

// Deterministic fixed-tree finalize: per-image sums -> /max(num_pos,1) -> mean
__global__ __launch_bounds__(TPB) void focal_finalize_kernel(
    const float* __restrict__ ws_cls, const float* __restrict__ ws_reg,
    const float* __restrict__ ws_np, float* __restrict__ out,
    int B, int nblk) {
  __shared__ float sc[TPB], sr[TPB], sn[TPB];
  float cm = 0.0f, rm = 0.0f;
  for (int b = 0; b < B; ++b) {
    float c = 0.0f, r = 0.0f, n = 0.0f;
    for (int i = threadIdx.x; i < nblk; i += TPB) {
      c += ws_cls[(size_t)b * nblk + i];
      r += ws_reg[(size_t)b * nblk + i];
      n += ws_np[(size_t)b * nblk + i];
    }
    sc[threadIdx.x] = c; sr[threadIdx.x] = r; sn[threadIdx.x] = n;
    __syncthreads();
    for (int s = TPB / 2; s > 0; s >>= 1) {
      if (threadIdx.x < s) {
        sc[threadIdx.x] += sc[threadIdx.x + s];
        sr[threadIdx.x] += sr[threadIdx.x + s];
        sn[threadIdx.x] += sn[threadIdx.x + s];
      }
      __syncthreads();
    }
    if (threadIdx.x == 0) {
      const float denom = fmaxf(sn[0], 1.0f);
      cm += sc[0] / denom;
      rm += sr[0] / denom;
    }
    __syncthreads();
  }
  if (threadIdx.x == 0) {
    out[0] = cm / (float)B;
    out[1] = rm / (float)B;
  }
}

extern "C" void kernel_launch(void* const* d_in, const int* in_sizes, int n_in,
                              void* d_out, int out_size, void* d_ws, size_t ws_size,
                              hipStream_t stream) {
  const float* cls_p = (const float*)d_in[0];  // (B, A, C)
  const float* regr  = (const float*)d_in[1];  // (B, A, 4)
  const float* anch  = (const float*)d_in[2];  // (1, A, 4)
  const float* ann   = (const float*)d_in[3];  // (B, M, 5)
  float* out = (float*)d_out;                  // (cls_mean, reg_mean)

  const int A = in_sizes[2] / 4;
  const int B = in_sizes[1] / (A * 4);
  const int C = in_sizes[0] / (B * A);
  const int M = in_sizes[3] / (B * 5);

  const int nblk = (A + TPB - 1) / TPB;
  float* ws_cls = (float*)d_ws;
  float* ws_reg = ws_cls + (size_t)B * nblk;
  float* ws_np  = ws_reg + (size_t)B * nblk;

  dim3 grid(nblk, B);
  if (C == 80 && M == 32) {
    focal_main_kernel<80, 32><<<grid, TPB, 0, stream>>>(
        cls_p, regr, anch, ann, ws_cls, ws_reg, ws_np, A, C, M);
  } else {
    focal_main_kernel<0, 0><<<grid, TPB, 0, stream>>>(
        cls_p, regr, anch, ann, ws_cls, ws_reg, ws_np, A, C, M);
  }
  focal_finalize_kernel<<<1, TPB, 0, stream>>>(ws_cls, ws_reg, ws_np, out, B, nblk);
}